// MPULoss_V2_1778116461028
// MI455X (gfx1250) — compile-verified
//
#include <hip/hip_runtime.h>
#include <math.h>

#define NUM_CLASS 1000
#define EPS 1e-6f
#define PU_W 2.0f
// logf(1.0f + 1e-6f) precomputed in f32
#define LOG1PEPS 9.9999950e-07f
#define NEG_BIG -3.402823466e+38f
#define LOG2E_F 1.44269504088896340736f
#define LN2_F 0.693147180559945309417f

typedef __attribute__((ext_vector_type(2))) float v2f;
typedef __attribute__((ext_vector_type(8))) float v8f;

// Raw transcendentals: arguments here are always normal/negative-bounded, so
// skip libm's denorm-guard prologues and use the HW ops directly.
__device__ __forceinline__ float fast_exp(float x) {          // e^x
  return __builtin_amdgcn_exp2f(x * LOG2E_F);                 // v_exp_f32
}
__device__ __forceinline__ float fast_log2(float x) {         // log2(x), x normal
  return __builtin_amdgcn_logf(x);                            // v_log_f32
}

// Wave-wide f32 sum using V_WMMA_F32_16X16X4_F32 with B = ones.
// A layout (16x4 f32): lanes 0-15 hold {K0,K1} for M=lane, lanes 16-31 hold
// {K2,K3} for M=lane-16. With a = {v, 0}, D[m,n] = v_m + v_{m+16} for all n.
// Lane L (0-15) holds D[M=0..7][L] in 8 VGPRs, lane L+16 holds M=8..15, so
// sum(c[0..7]) + shfl_xor(.,16) = exact f32 sum over all 32 lanes.
__device__ __forceinline__ float wave_sum_wmma(float v) {
  v2f a; a[0] = v;    a[1] = 0.0f;
  v2f b; b[0] = 1.0f; b[1] = 1.0f;
  v8f c = {0.f, 0.f, 0.f, 0.f, 0.f, 0.f, 0.f, 0.f};
  c = __builtin_amdgcn_wmma_f32_16x16x4_f32(false, a, false, b, (short)0, c,
                                            false, false);
  float s = ((c[0] + c[1]) + (c[2] + c[3])) + ((c[4] + c[5]) + (c[6] + c[7]));
  s += __shfl_xor(s, 16, 32);
  return s;
}

__device__ __forceinline__ float wave_max(float v) {
#pragma unroll
  for (int m = 16; m >= 1; m >>= 1) v = fmaxf(v, __shfl_xor(v, m, 32));
  return v;
}

// One wave per row. Lane owns columns c = j*128 + lane*4 + k, j=0..7, k=0..3
// (chunk 7 valid only for lanes < 26 : 896 + 26*4 = 1000 exactly).
__global__ __launch_bounds__(256) void pu_row_kernel(
    const float* __restrict__ outputs, const int* __restrict__ labels,
    const float* __restrict__ prior, const int* __restrict__ idxlist,
    float* __restrict__ partials, int nRows, int nBlk) {
  __shared__ __align__(16) float sIdx[1024];
  __shared__ float sRed[8][4];

  const int lane = threadIdx.x & 31;
  const int wave = threadIdx.x >> 5;
  const int row  = blockIdx.x * 8 + wave;
  const bool active = (row < nRows);

  // Kick the row into cache early (global_prefetch_b8), one line per lane.
  if (active) {
    const char* p = (const char*)(outputs + (size_t)row * NUM_CLASS);
    __builtin_prefetch(p + lane * 128, 0, 0);
  }

  // Stage class mask (as float) in LDS once per block.
  for (int i = threadIdx.x; i < 1024; i += 256)
    sIdx[i] = (i < NUM_CLASS) ? (float)idxlist[i] : 0.0f;
  __syncthreads();

  float r0 = 0.f, r1 = 0.f, r2 = 0.f, r3 = 0.f;

  if (active) {
    const float4* rowp = (const float4*)(outputs + (size_t)row * NUM_CLASS);
    float4 xv[8];
#pragma unroll
    for (int j = 0; j < 7; ++j) xv[j] = rowp[j * 32 + lane];
    {
      float4 t;
      if (lane < 26) t = rowp[224 + lane];
      else { t.x = NEG_BIG; t.y = NEG_BIG; t.z = NEG_BIG; t.w = NEG_BIG; }
      xv[7] = t;
    }

    // Row max
    float mx = NEG_BIG;
#pragma unroll
    for (int j = 0; j < 8; ++j)
      mx = fmaxf(mx, fmaxf(fmaxf(xv[j].x, xv[j].y), fmaxf(xv[j].z, xv[j].w)));
    mx = wave_max(mx);

    // Pass B: e = exp(x - mx) (stored back into xv), Z partial sum.
    float zAcc = 0.f;
#pragma unroll
    for (int j = 0; j < 8; ++j) {
      if (j == 7 && lane >= 26) continue;
      float4 x = xv[j];
      float4 e;
      e.x = fast_exp(x.x - mx); e.y = fast_exp(x.y - mx);
      e.z = fast_exp(x.z - mx); e.w = fast_exp(x.w - mx);
      zAcc += (e.x + e.y) + (e.z + e.w);
      xv[j] = e;
    }
    const float Z    = wave_sum_wmma(zAcc);
    const float invZ = __builtin_amdgcn_rcpf(Z);       // v_rcp_f32
    const float logZ = fast_log2(Z) * LN2_F;           // Z >= 1, always normal

    // Pass C: term partial sum (in log2 units; one ln2 scale at the end).
    const float4* sIdx4 = (const float4*)sIdx;
    float tAcc = 0.f;
#pragma unroll
    for (int j = 0; j < 8; ++j) {
      if (j == 7 && lane >= 26) continue;
      float4 e  = xv[j];
      float4 iv = sIdx4[j * 32 + lane];
      // 1 - p + eps >= eps = 1e-6 : always a normal float, raw v_log_f32 OK.
      tAcc -= fast_log2(1.0f - e.x * invZ + EPS) * iv.x;
      tAcc -= fast_log2(1.0f - e.y * invZ + EPS) * iv.y;
      tAcc -= fast_log2(1.0f - e.z * invZ + EPS) * iv.z;
      tAcc -= fast_log2(1.0f - e.w * invZ + EPS) * iv.w;
    }
    const float term = wave_sum_wmma(tAcc) * LN2_F;

    const int lab = labels[row];
    if (lab < NUM_CLASS) {  // labeled row: re-read x[lab] (L0-hot, uniform)
      const float xLab = outputs[(size_t)row * NUM_CLASS + lab];
      const float pLab = fast_exp(xLab - mx) * invZ;
      const float plab = prior[lab];
      // per-row variable part of pu2: prior[lab]*(-log(1-p+eps) + log(1+eps))
      r1 = plab * (-(fast_log2(1.0f - pLab + EPS) * LN2_F) + LOG1PEPS);
      r2 = mx + logZ - xLab;  // cross-entropy term
      r3 = 1.0f;              // nP count
    } else {                  // unlabeled row
      r0 = term;
    }
  }

  if (lane == 0) {
    sRed[wave][0] = r0; sRed[wave][1] = r1;
    sRed[wave][2] = r2; sRed[wave][3] = r3;
  }
  __syncthreads();
  if (threadIdx.x < 4) {
    float s = 0.f;
#pragma unroll
    for (int w = 0; w < 8; ++w) s += sRed[w][threadIdx.x];
    partials[threadIdx.x * nBlk + blockIdx.x] = s;  // overwrite: no ws init needed
  }
}

__global__ __launch_bounds__(256) void pu_finalize_kernel(
    const float* __restrict__ partials, const float* __restrict__ prior,
    float* __restrict__ out, int nBlk, int nRows) {
  float a0 = 0.f, a1 = 0.f, a2 = 0.f, a3 = 0.f, aP = 0.f;
  for (int i = threadIdx.x; i < nBlk; i += 256) {
    a0 += partials[i];
    a1 += partials[nBlk + i];
    a2 += partials[2 * nBlk + i];
    a3 += partials[3 * nBlk + i];
  }
  for (int i = threadIdx.x; i < NUM_CLASS; i += 256) aP += prior[i];

  a0 = wave_sum_wmma(a0); a1 = wave_sum_wmma(a1); a2 = wave_sum_wmma(a2);
  a3 = wave_sum_wmma(a3); aP = wave_sum_wmma(aP);

  __shared__ float red[8][5];
  const int lane = threadIdx.x & 31, w = threadIdx.x >> 5;
  if (lane == 0) {
    red[w][0] = a0; red[w][1] = a1; red[w][2] = a2; red[w][3] = a3; red[w][4] = aP;
  }
  __syncthreads();
  if (threadIdx.x == 0) {
    float S0 = 0.f, S1 = 0.f, S2 = 0.f, S3 = 0.f, SP = 0.f;
#pragma unroll
    for (int i = 0; i < 8; ++i) {
      S0 += red[i][0]; S1 += red[i][1]; S2 += red[i][2];
      S3 += red[i][3]; SP += red[i][4];
    }
    const float nP = S3;
    const float nU = (float)nRows - nP;
    const float pu3 = S0 / fmaxf(1.0f, nU) / (float)NUM_CLASS;
    // pu2 = -( sum_labeled [ -log(1+eps)*sumPrior + var-part ] ) / max(1,nP)
    const float total = (-LOG1PEPS) * SP * nP + S1;
    const float pu2 = -total / fmaxf(1.0f, nP);
    const float PULoss = pu3 + pu2;
    const float cross = S2 / nP;  // 0/0 -> NaN when nP == 0, matching reference
    const float PULossW = PULoss * PU_W;
    const float obj = (cross != cross) ? PULoss : (PULossW + cross);
    out[0] = obj; out[1] = PULossW; out[2] = cross;
  }
}

extern "C" void kernel_launch(void* const* d_in, const int* in_sizes, int n_in,
                              void* d_out, int out_size, void* d_ws, size_t ws_size,
                              hipStream_t stream) {
  const float* outputs = (const float*)d_in[0];
  const int*   labels  = (const int*)d_in[1];
  const float* prior   = (const float*)d_in[2];
  const int*   idxlist = (const int*)d_in[3];
  float* out      = (float*)d_out;
  float* partials = (float*)d_ws;  // needs 4 * nBlk * 4 bytes = 128 KB

  const int nRows = in_sizes[1];          // 65536 labels -> 65536 rows
  const int nBlk  = (nRows + 7) / 8;      // 8 rows (waves) per block

  pu_row_kernel<<<nBlk, 256, 0, stream>>>(outputs, labels, prior, idxlist,
                                          partials, nRows, nBlk);
  pu_finalize_kernel<<<1, 256, 0, stream>>>(partials, prior, out, nBlk, nRows);
}